// INOUnit_48241072669164
// MI455X (gfx1250) — compile-verified
//
#include <hip/hip_runtime.h>
#include <hip/hip_bf16.h>

#define DIM 128

typedef __attribute__((ext_vector_type(2))) float v2f;
typedef __attribute__((ext_vector_type(8))) float v8f;

// ---------------------------------------------------------------------------
// Degree / normalization kernels
// ---------------------------------------------------------------------------
__global__ void deg_init_kernel(unsigned* __restrict__ degu, int n) {
    int i = blockIdx.x * blockDim.x + threadIdx.x;
    if (i < n) degu[i] = 1u;  // self loop
}

__global__ void deg_count_kernel(const int* __restrict__ dst, unsigned* __restrict__ degu, int e) {
    int i = blockIdx.x * blockDim.x + threadIdx.x;
    if (i < e) atomicAdd(&degu[dst[i]], 1u);
}

__global__ void dis_kernel(const unsigned* __restrict__ degu, float* __restrict__ dis, int n) {
    int i = blockIdx.x * blockDim.x + threadIdx.x;
    if (i < n) dis[i] = rsqrtf((float)degu[i]);
}

// ---------------------------------------------------------------------------
// Dense GEMM  H = X @ W  using V_WMMA_F32_16X16X4_F32 (full fp32 precision).
// One block = 128 rows of X; W (128x128, 64KB) staged in LDS; 8 waves each
// own a 16-row strip and 8 column tiles (8 x v8f accumulators).
// ---------------------------------------------------------------------------
__global__ void __launch_bounds__(256) gemm_wmma_f32_kernel(
    const float* __restrict__ X, const float* __restrict__ W,
    float* __restrict__ H, int n_rows) {
    __shared__ float Wl[DIM * DIM];  // [k][n], row-major

    // cooperative load of W into LDS (float4)
    {
        const float4* Wv = (const float4*)W;
        float4* Wlv = (float4*)Wl;
        for (int i = threadIdx.x; i < DIM * DIM / 4; i += blockDim.x) Wlv[i] = Wv[i];
    }
    __syncthreads();

    const int wave = threadIdx.x >> 5;
    const int lane = threadIdx.x & 31;
    const int mrow0 = blockIdx.x * 128 + wave * 16;
    const int ml = lane & 15;           // row within tile (A) / col within tile (B)
    const int khalf = (lane >> 4) << 1; // lanes 0-15 -> K {0,1}, lanes 16-31 -> K {2,3}

    int arow = mrow0 + ml;
    if (arow >= n_rows) arow = n_rows - 1;  // clamp (stores are guarded below)
    const float* Arow = X + (size_t)arow * DIM;

    v8f acc[8] = {};

    for (int k0 = 0; k0 < DIM; k0 += 4) {
        const int ka = k0 + khalf;
        v2f a;
        a.x = Arow[ka];
        a.y = Arow[ka + 1];
#pragma unroll
        for (int c = 0; c < 8; ++c) {
            const int n = c * 16 + ml;
            v2f b;
            b.x = Wl[ka * DIM + n];
            b.y = Wl[(ka + 1) * DIM + n];
            acc[c] = __builtin_amdgcn_wmma_f32_16x16x4_f32(
                /*neg_a=*/false, a, /*neg_b=*/false, b,
                /*c_mod=*/(short)0, acc[c], /*reuse_a=*/false, /*reuse_b=*/false);
        }
    }

    // C/D layout: lanes 0-15 hold M = r (VGPR r), lanes 16-31 hold M = 8 + r
    const int mtop = mrow0 + ((lane >> 4) << 3);
    if (mrow0 + 16 <= n_rows) {
        // Fast path (all blocks except the last partial one): unguarded,
        // fully coalesced stores — no per-store exec-mask gymnastics.
#pragma unroll
        for (int c = 0; c < 8; ++c) {
#pragma unroll
            for (int r = 0; r < 8; ++r) {
                H[(size_t)(mtop + r) * DIM + c * 16 + ml] = acc[c][r];
            }
        }
    } else {
        // Tail path: per-row guard.
#pragma unroll
        for (int c = 0; c < 8; ++c) {
#pragma unroll
            for (int r = 0; r < 8; ++r) {
                const int m = mtop + r;
                if (m < n_rows) H[(size_t)m * DIM + c * 16 + ml] = acc[c][r];
            }
        }
    }
}

// ---------------------------------------------------------------------------
// Aggregation: init with bias + self-loop term (norm = dis[n]^2)
// ---------------------------------------------------------------------------
__global__ void agg_init_kernel(const float* __restrict__ h, const float* __restrict__ dis,
                                const float* __restrict__ bias, float* __restrict__ out, int n) {
    int i = blockIdx.x * blockDim.x + threadIdx.x;
    if (i < n * DIM) {
        int node = i >> 7;
        int f = i & (DIM - 1);
        float w = dis[node];
        out[i] = bias[f] + w * w * h[i];
    }
}

// Edge scatter: one wave32 per edge, each lane gathers float4 from h[src],
// scales by dis[src]*dis[dst], and does 4 f32 atomic adds into out[dst].
__global__ void scatter_kernel(const float* __restrict__ h, const float* __restrict__ dis,
                               const int* __restrict__ src, const int* __restrict__ dst,
                               float* __restrict__ out, int e) {
    int edge = blockIdx.x * (blockDim.x >> 5) + (threadIdx.x >> 5);
    int lane = threadIdx.x & 31;
    if (edge >= e) return;
    int s = src[edge];
    int d = dst[edge];
    float nrm = dis[s] * dis[d];
    float4 v = ((const float4*)(h + (size_t)s * DIM))[lane];
    float* o = out + (size_t)d * DIM + lane * 4;
    atomicAdd(o + 0, v.x * nrm);
    atomicAdd(o + 1, v.y * nrm);
    atomicAdd(o + 2, v.z * nrm);
    atomicAdd(o + 3, v.w * nrm);
}

// ---------------------------------------------------------------------------
// Coupling combine: out = xbase * exp(tanh(s_raw)) + out   (out holds t_raw)
// ---------------------------------------------------------------------------
__global__ void combine_kernel(const float* __restrict__ xbase, const float* __restrict__ s_raw,
                               float* __restrict__ out, int total) {
    int i = blockIdx.x * blockDim.x + threadIdx.x;
    if (i < total) out[i] = xbase[i] * __expf(tanhf(s_raw[i])) + out[i];
}

// ---------------------------------------------------------------------------
extern "C" void kernel_launch(void* const* d_in, const int* in_sizes, int n_in,
                              void* d_out, int out_size, void* d_ws, size_t ws_size,
                              hipStream_t stream) {
    const float* x_main  = (const float*)d_in[0];
    const float* x_guide = (const float*)d_in[1];
    const int*   eidx    = (const int*)d_in[2];
    const float* W_s1 = (const float*)d_in[3];
    const float* b_s1 = (const float*)d_in[4];
    const float* W_t1 = (const float*)d_in[5];
    const float* b_t1 = (const float*)d_in[6];
    const float* W_s2 = (const float*)d_in[7];
    const float* b_s2 = (const float*)d_in[8];
    const float* W_t2 = (const float*)d_in[9];
    const float* b_t2 = (const float*)d_in[10];

    const int N = in_sizes[0] / DIM;
    const int E = in_sizes[2] / 2;
    const int* src = eidx;       // edge_index row 0
    const int* dst = eidx + E;   // edge_index row 1

    float* out_main  = (float*)d_out;
    float* out_guide = out_main + (size_t)N * DIM;

    char* ws = (char*)d_ws;
    float*    dis  = (float*)ws;                           // N floats
    unsigned* degu = (unsigned*)(ws + (size_t)N * 4);      // N uints
    float*    h    = (float*)(ws + (size_t)N * 8);         // N*128 floats (16B aligned)
    float*    agg  = h + (size_t)N * DIM;                  // N*128 floats

    const int nblk  = (N + 255) / 256;
    const int eblk  = (E + 255) / 256;
    const int gemmg = (N + 127) / 128;
    const int total = N * DIM;
    const int elblk = (total + 255) / 256;
    const int scblk = (E + 7) / 8;

    // normalization
    deg_init_kernel<<<nblk, 256, 0, stream>>>(degu, N);
    deg_count_kernel<<<eblk, 256, 0, stream>>>(dst, degu, E);
    dis_kernel<<<nblk, 256, 0, stream>>>(degu, dis, N);

    // --- stage 1: t1 directly into out_main, s1 into agg ---
    gemm_wmma_f32_kernel<<<gemmg, 256, 0, stream>>>(x_guide, W_t1, h, N);
    agg_init_kernel<<<elblk, 256, 0, stream>>>(h, dis, b_t1, out_main, N);
    scatter_kernel<<<scblk, 256, 0, stream>>>(h, dis, src, dst, out_main, E);

    gemm_wmma_f32_kernel<<<gemmg, 256, 0, stream>>>(x_guide, W_s1, h, N);
    agg_init_kernel<<<elblk, 256, 0, stream>>>(h, dis, b_s1, agg, N);
    scatter_kernel<<<scblk, 256, 0, stream>>>(h, dis, src, dst, agg, E);

    combine_kernel<<<elblk, 256, 0, stream>>>(x_main, agg, out_main, total);

    // --- stage 2: t2 into out_guide, s2 into agg (inputs = out_main) ---
    gemm_wmma_f32_kernel<<<gemmg, 256, 0, stream>>>(out_main, W_t2, h, N);
    agg_init_kernel<<<elblk, 256, 0, stream>>>(h, dis, b_t2, out_guide, N);
    scatter_kernel<<<scblk, 256, 0, stream>>>(h, dis, src, dst, out_guide, E);

    gemm_wmma_f32_kernel<<<gemmg, 256, 0, stream>>>(out_main, W_s2, h, N);
    agg_init_kernel<<<elblk, 256, 0, stream>>>(h, dis, b_s2, agg, N);
    scatter_kernel<<<scblk, 256, 0, stream>>>(h, dis, src, dst, agg, E);

    combine_kernel<<<elblk, 256, 0, stream>>>(x_guide, agg, out_guide, total);
}